// All2AllDenseEmbedding_28827820491521
// MI455X (gfx1250) — compile-verified
//
#include <hip/hip_runtime.h>
#include <hip/hip_bf16.h>

// Embedding gather: out[r, :] = table[idx[r], :]  (row = 32 f32 = 128 bytes)
// 8 lanes per row, one 16B chunk (B128) per lane. Pure bandwidth kernel:
// ~56 MB total traffic -> ~2.4 us at 23.3 TB/s. Table (128 MB) fits in the
// 192 MB L2, so duplicate indices hit cache; output is streamed with NT hint.

using i64 = long long;
typedef __attribute__((ext_vector_type(4))) float v4f;  // native vec for NT builtins

__global__ __launch_bounds__(256) void All2AllDenseEmbedding_gather_kernel(
    const i64* __restrict__ idxs,   // [nrows] int64 row indices
    const v4f* __restrict__ tab4,   // table viewed as [vocab*8] v4f
    v4f* __restrict__ out4,         // output viewed as [nrows*8] v4f
    int nrows)
{
    const int t     = blockIdx.x * 256 + threadIdx.x;
    const int total = nrows << 3;              // 8 x 16B chunks per row
    if (t >= total) return;

    const int row   = t >> 3;                  // which embedding row
    const int chunk = t & 7;                   // which 16B chunk of the row

    // Speculative prefetch of the index stream ~64KB ahead
    // (gfx1250 global_prefetch_b8; OOB translations are silently dropped).
    __builtin_prefetch(idxs + row + 8192, 0, 0);

    // 8 lanes share one 8B index -> coalesces to a single 32B request/wave.
    const i64 idx = idxs[row];

    // Offset fits 32 bits: idx < 1e6, idx*8 chunks < 2^23.
    const unsigned off = (unsigned)idx * 8u + (unsigned)chunk;

    // RT-hint B128 load (wants L2 residency for repeated rows).
    const v4f v = tab4[off];

    // Streaming B128 store with non-temporal hint: written once, never reread.
    __builtin_nontemporal_store(v, out4 + t);
}

extern "C" void kernel_launch(void* const* d_in, const int* in_sizes, int n_in,
                              void* d_out, int out_size, void* d_ws, size_t ws_size,
                              hipStream_t stream) {
    const i64* idxs = (const i64*)d_in[0];     // [4096*26*2] int64
    const v4f* tab4 = (const v4f*)d_in[1];     // [1e6 * 32] f32 as v4f
    v4f*       out4 = (v4f*)d_out;             // [nrows * 32] f32 as v4f

    const int nrows = in_sizes[0];             // 212,992
    const int total = nrows << 3;              // one thread per 16B chunk
    const int block = 256;                     // 8 wave32s per block
    const int grid  = (total + block - 1) / block; // 6656 blocks

    All2AllDenseEmbedding_gather_kernel<<<grid, block, 0, stream>>>(
        idxs, tab4, out4, nrows);
}